// EdgeDecoder_60730837565672
// MI455X (gfx1250) — compile-verified
//
#include <hip/hip_runtime.h>

// Problem constants from the reference: B=8, N=4096, H=256.
#define B_ 8
#define N_ 4096
#define H_ 256
#define TJ 16   // 16x16 j-tiles per wave in the outer-sum kernel

typedef __attribute__((ext_vector_type(2))) float v2f;
typedef __attribute__((ext_vector_type(8))) float v8f;

// ---------------------------------------------------------------------------
// Phase 1: row[b,n] = z[b,n,:] . w[0:H] + bias ; col[b,n] = z[b,n,:] . w[H:2H]
// One wave per (b,n) row; each lane covers 8 of the 256 elements via 2x float4,
// then a 5-step wave32 xor-shuffle reduction.
// ---------------------------------------------------------------------------
__global__ __launch_bounds__(256) void rowcol_kernel(
    const float* __restrict__ z, const float* __restrict__ w,
    const float* __restrict__ bias,
    float* __restrict__ rowp, float* __restrict__ colp)
{
    const int lane = threadIdx.x & 31;
    const int wid  = threadIdx.x >> 5;
    const int row  = blockIdx.x * 8 + wid;        // 0 .. B*N-1

    const float4* zr = (const float4*)(z + (size_t)row * H_);
    const float4* wr = (const float4*)(w);        // first half of w
    const float4* wc = (const float4*)(w + H_);   // second half of w

    float4 z0 = zr[lane * 2 + 0];
    float4 z1 = zr[lane * 2 + 1];
    float4 r0 = wr[lane * 2 + 0];
    float4 r1 = wr[lane * 2 + 1];
    float4 c0 = wc[lane * 2 + 0];
    float4 c1 = wc[lane * 2 + 1];

    float sr = z0.x * r0.x + z0.y * r0.y + z0.z * r0.z + z0.w * r0.w
             + z1.x * r1.x + z1.y * r1.y + z1.z * r1.z + z1.w * r1.w;
    float sc = z0.x * c0.x + z0.y * c0.y + z0.z * c0.z + z0.w * c0.w
             + z1.x * c1.x + z1.y * c1.y + z1.z * c1.z + z1.w * c1.w;

    #pragma unroll
    for (int off = 16; off >= 1; off >>= 1) {
        sr += __shfl_xor(sr, off, 32);
        sc += __shfl_xor(sc, off, 32);
    }

    if (lane == 0) {
        rowp[row] = sr + bias[0];   // fold bias into the row term
        colp[row] = sc;
    }
}

// ---------------------------------------------------------------------------
// Phase 2: out[b,i,j] = row[b,i] + col[b,j], computed as a rank-2 WMMA:
//   A (16x4):  K0 = row tile, K1 = 1, K2 = 1, K3 = 0   (K1/K2 both 1 so the
//              result is correct for either candidate B-matrix K layout)
//   B (4x16):  V0.lo16 = 1 (K0), V0.hi16 = col tile (K1 or K2), V1 = 0
//   D = A x B  -> D[i,j] = row[i]*1 + 1*col[j]
// Each wave owns one 16-row i-tile and TJ consecutive 16-col j-tiles.
// D layout: lane l, vgpr r -> M = r + 8*(l>=16), N = l&15 -> 8 NT b32 stores.
// ---------------------------------------------------------------------------
__global__ __launch_bounds__(256) void outer_sum_wmma_kernel(
    const float* __restrict__ rowp, const float* __restrict__ colp,
    float* __restrict__ out)
{
    const int lane = threadIdx.x & 31;
    const int wid  = threadIdx.x >> 5;
    const int g    = blockIdx.x * 8 + wid;        // global wave id

    const int jc = g & (N_ / (16 * TJ) - 1);      // 16 j-chunks
    const int it = (g >> 4) & (N_ / 16 - 1);      // 256 i-tiles
    const int b  = g >> 12;                       // batch

    const int  nl = lane & 15;
    const bool hi = lane >= 16;
    const int  i0 = it * 16;
    const int  j0 = jc * (16 * TJ);

    // A-matrix (documented 16x4 f32 layout: V0={K0 lo | K2 hi}, V1={K1 lo | K3 hi})
    const float rv = rowp[b * N_ + i0 + nl];
    v2f a;
    a.x = hi ? 1.0f : rv;     // lo: K0 = row ; hi: K2 = 1
    a.y = hi ? 0.0f : 1.0f;   // lo: K1 = 1   ; hi: K3 = 0

    const float* colb = colp + b * N_ + j0;
    size_t outbase = (((size_t)b * N_) + i0 + (hi ? 8 : 0)) * N_ + j0 + nl;

    for (int t = 0; t < TJ; ++t) {
        const float cv = colb[t * 16 + nl];
        v2f bm;
        bm.x = hi ? cv : 1.0f;    // lo: K0 = 1 ; hi: col (K1 or K2 — both safe)
        bm.y = 0.0f;              // remaining K slots zero
        v8f c = {};
        v8f d = __builtin_amdgcn_wmma_f32_16x16x4_f32(
            false, a, false, bm, (short)0, c, false, false);

        float* o = out + outbase + (size_t)t * 16;
        #pragma unroll
        for (int r = 0; r < 8; ++r) {
            __builtin_nontemporal_store(d[r], o + (size_t)r * N_);
        }
    }
}

extern "C" void kernel_launch(void* const* d_in, const int* in_sizes, int n_in,
                              void* d_out, int out_size, void* d_ws, size_t ws_size,
                              hipStream_t stream) {
    const float* z    = (const float*)d_in[0];   // [B, N, H] f32
    const float* w    = (const float*)d_in[1];   // [2H] f32
    const float* bias = (const float*)d_in[2];   // [1] f32
    float* out = (float*)d_out;                  // [B, N, N] f32

    float* rowp = (float*)d_ws;                  // B*N floats
    float* colp = rowp + (size_t)B_ * N_;        // B*N floats (total 256 KiB)

    // Phase 1: 32768 rows, 1 wave per row, 8 waves per block.
    rowcol_kernel<<<(B_ * N_) / 8, 256, 0, stream>>>(z, w, bias, rowp, colp);

    // Phase 2: B * (N/16) * (N/(16*TJ)) = 32768 waves, 8 waves per block.
    const int waves = B_ * (N_ / 16) * (N_ / (16 * TJ));
    outer_sum_wmma_kernel<<<waves / 8, 256, 0, stream>>>(rowp, colp, out);
}